// TransformerBlock_68332929679844
// MI455X (gfx1250) — compile-verified
//
#include <hip/hip_runtime.h>
#include <math.h>

typedef __attribute__((ext_vector_type(2))) float v2f;
typedef __attribute__((ext_vector_type(8))) float v8f;

#define D_MODEL 1024
#define N_HEADS 16
#define D_K     64
#define D_FF    4096
#define SEQ     2048
#define BATCH   2
#define M_TOT   (BATCH * SEQ)                // 4096 rows
#define BUF_ELEMS ((size_t)M_TOT * D_MODEL)  // 4,194,304 floats

// ---- fp32 WMMA: D(16x16) = A(16x4) * B(4x16) + C ----------------------------
__device__ __forceinline__ v8f wmma4(v2f a, v2f b, v8f c) {
    // 8-arg form: (neg_a, A, neg_b, B, c_mod, C, reuse_a, reuse_b)
    return __builtin_amdgcn_wmma_f32_16x16x4_f32(false, a, false, b, (short)0, c,
                                                 false, false);
}

// ---- RMSNorm: one 256-thread block per row of 1024 --------------------------
__global__ void rmsnorm_kernel(const float* __restrict__ x,
                               const float* __restrict__ w,
                               float* __restrict__ out) {
    int row = blockIdx.x;
    const float* xr = x + (size_t)row * D_MODEL;
    float* orow = out + (size_t)row * D_MODEL;
    int t = threadIdx.x;                       // 256 threads * float4 = 1024
    float4 v = ((const float4*)xr)[t];
    float ss = v.x * v.x + v.y * v.y + v.z * v.z + v.w * v.w;
#pragma unroll
    for (int m = 16; m >= 1; m >>= 1) ss += __shfl_xor(ss, m, 32);
    __shared__ float red[8];
    int wave = t >> 5, lane = t & 31;
    if (lane == 0) red[wave] = ss;
    __syncthreads();
    float tot = red[0] + red[1] + red[2] + red[3] +
                red[4] + red[5] + red[6] + red[7];
    float scale = rsqrtf(tot * (1.0f / (float)D_MODEL) + 1e-5f);
    float4 wv = ((const float4*)w)[t];
    float4 o;
    o.x = v.x * scale * wv.x;
    o.y = v.y * scale * wv.y;
    o.z = v.z * scale * wv.z;
    o.w = v.w * scale * wv.w;
    ((float4*)orow)[t] = o;
}

// ---- RoPE (in place on ws buffer), layout (b, s, h*64+d) --------------------
__global__ void rope_kernel(float* __restrict__ x, int npairs) {
    int idx = blockIdx.x * blockDim.x + threadIdx.x;
    if (idx >= npairs) return;
    int colpair = idx & (D_MODEL / 2 - 1);     // e/2
    int row = idx / (D_MODEL / 2);
    int s = row & (SEQ - 1);
    int p = colpair & (D_K / 2 - 1);           // pair index within head
    // inv_freq = theta^(-2p/64), log(10000) = 9.210340371976184
    float inv_freq = __expf(-(float)(2 * p) * (1.0f / (float)D_K) * 9.210340371976184f);
    float ang = (float)s * inv_freq;
    float sn, cs;
    __sincosf(ang, &sn, &cs);
    float* pr = x + (size_t)row * D_MODEL + 2 * colpair;
    float e = pr[0], o = pr[1];
    pr[0] = e * cs - o * sn;
    pr[1] = e * sn + o * cs;
}

// ---- GEMM NT: C[M,N] = A[M,K] @ B[N,K]^T (+ residual when MODE==1) ----------
// 8 waves/block in a 2(m) x 4(n) grid; each wave computes a 32x64 tile
// (2 m-tiles x 4 n-tiles = 8 accumulators). Block tile: 64(M) x 256(N).
// Per k-step: 6 b64 loads feed 8 WMMAs (0.75 loads/wmma), software-pipelined.
template <int MODE>
__global__ void gemm_nt_kernel(const float* __restrict__ A,
                               const float* __restrict__ B,
                               float* __restrict__ C,
                               const float* __restrict__ R,
                               int M, int N, int K) {
    int wave = threadIdx.x >> 5;
    int lane = threadIdx.x & 31;
    int hi = lane >> 4, lo = lane & 15;
    int wave_m = wave & 1;          // 0..1
    int wave_n = wave >> 1;         // 0..3
    int mbase = blockIdx.x * 64 + wave_m * 32;   // wave covers rows [mbase, mbase+32)
    int nbase = blockIdx.y * 256 + wave_n * 64;  // wave covers cols [nbase, nbase+64)
    if (mbase >= M || nbase >= N) return;

    v8f acc00 = {}, acc01 = {}, acc02 = {}, acc03 = {};
    v8f acc10 = {}, acc11 = {}, acc12 = {}, acc13 = {};

    const float* ar0 = A + (size_t)(mbase + lo) * K + 2 * hi;
    const float* ar1 = ar0 + (size_t)16 * K;
    const float* br0 = B + (size_t)(nbase + lo) * K + 2 * hi;
    const float* br1 = br0 + (size_t)16 * K;
    const float* br2 = br0 + (size_t)32 * K;
    const float* br3 = br0 + (size_t)48 * K;

    // software pipeline: fragments for step k resident while k+4 loads in flight
    v2f a0 = *(const v2f*)(ar0);
    v2f a1 = *(const v2f*)(ar1);
    v2f b0 = *(const v2f*)(br0);
    v2f b1 = *(const v2f*)(br1);
    v2f b2 = *(const v2f*)(br2);
    v2f b3 = *(const v2f*)(br3);

    for (int k0 = 0; k0 < K - 4; k0 += 4) {
        int kn = k0 + 4;
        v2f na0 = *(const v2f*)(ar0 + kn);
        v2f na1 = *(const v2f*)(ar1 + kn);
        v2f nb0 = *(const v2f*)(br0 + kn);
        v2f nb1 = *(const v2f*)(br1 + kn);
        v2f nb2 = *(const v2f*)(br2 + kn);
        v2f nb3 = *(const v2f*)(br3 + kn);
        acc00 = wmma4(a0, b0, acc00);
        acc01 = wmma4(a0, b1, acc01);
        acc02 = wmma4(a0, b2, acc02);
        acc03 = wmma4(a0, b3, acc03);
        acc10 = wmma4(a1, b0, acc10);
        acc11 = wmma4(a1, b1, acc11);
        acc12 = wmma4(a1, b2, acc12);
        acc13 = wmma4(a1, b3, acc13);
        a0 = na0; a1 = na1;
        b0 = nb0; b1 = nb1; b2 = nb2; b3 = nb3;
    }
    acc00 = wmma4(a0, b0, acc00);
    acc01 = wmma4(a0, b1, acc01);
    acc02 = wmma4(a0, b2, acc02);
    acc03 = wmma4(a0, b3, acc03);
    acc10 = wmma4(a1, b0, acc10);
    acc11 = wmma4(a1, b1, acc11);
    acc12 = wmma4(a1, b2, acc12);
    acc13 = wmma4(a1, b3, acc13);

#pragma unroll
    for (int v = 0; v < 8; ++v) {
        int r0 = mbase + v + 8 * hi;          // m-tile 0
        int r1 = r0 + 16;                      // m-tile 1
        size_t base0 = (size_t)r0 * N + nbase + lo;
        size_t base1 = (size_t)r1 * N + nbase + lo;
        float o00 = acc00[v], o01 = acc01[v], o02 = acc02[v], o03 = acc03[v];
        float o10 = acc10[v], o11 = acc11[v], o12 = acc12[v], o13 = acc13[v];
        if (MODE == 1) {
            o00 += R[base0];      o01 += R[base0 + 16];
            o02 += R[base0 + 32]; o03 += R[base0 + 48];
            o10 += R[base1];      o11 += R[base1 + 16];
            o12 += R[base1 + 32]; o13 += R[base1 + 48];
        }
        C[base0]      = o00; C[base0 + 16] = o01;
        C[base0 + 32] = o02; C[base0 + 48] = o03;
        C[base1]      = o10; C[base1 + 16] = o11;
        C[base1 + 32] = o12; C[base1 + 48] = o13;
    }
}

// ---- Fused FFN up-projection: H = silu(X@W1^T) * (X@W3^T) -------------------
// 8 waves/block in 2(m) x 4(n); each wave: 2 m-tiles x 2 n-tiles, computing
// both the w1 and w3 products (8 accumulators). Block tile: 64(M) x 128(N).
// Per k-step: 6 b64 loads feed 8 WMMAs.
__global__ void gemm_swiglu_kernel(const float* __restrict__ X,
                                   const float* __restrict__ W1,
                                   const float* __restrict__ W3,
                                   float* __restrict__ H,
                                   int M, int N, int K) {
    int wave = threadIdx.x >> 5;
    int lane = threadIdx.x & 31;
    int hi = lane >> 4, lo = lane & 15;
    int wave_m = wave & 1;
    int wave_n = wave >> 1;
    int mbase = blockIdx.x * 64 + wave_m * 32;
    int nbase = blockIdx.y * 128 + wave_n * 32;
    if (mbase >= M || nbase >= N) return;

    v8f g00 = {}, g01 = {}, g10 = {}, g11 = {};   // X@W1^T tiles
    v8f u00 = {}, u01 = {}, u10 = {}, u11 = {};   // X@W3^T tiles

    const float* ar0 = X + (size_t)(mbase + lo) * K + 2 * hi;
    const float* ar1 = ar0 + (size_t)16 * K;
    const float* w1r0 = W1 + (size_t)(nbase + lo) * K + 2 * hi;
    const float* w1r1 = w1r0 + (size_t)16 * K;
    const float* w3r0 = W3 + (size_t)(nbase + lo) * K + 2 * hi;
    const float* w3r1 = w3r0 + (size_t)16 * K;

    v2f a0 = *(const v2f*)(ar0);
    v2f a1 = *(const v2f*)(ar1);
    v2f p0 = *(const v2f*)(w1r0);
    v2f p1 = *(const v2f*)(w1r1);
    v2f q0 = *(const v2f*)(w3r0);
    v2f q1 = *(const v2f*)(w3r1);

    for (int k0 = 0; k0 < K - 4; k0 += 4) {
        int kn = k0 + 4;
        v2f na0 = *(const v2f*)(ar0 + kn);
        v2f na1 = *(const v2f*)(ar1 + kn);
        v2f np0 = *(const v2f*)(w1r0 + kn);
        v2f np1 = *(const v2f*)(w1r1 + kn);
        v2f nq0 = *(const v2f*)(w3r0 + kn);
        v2f nq1 = *(const v2f*)(w3r1 + kn);
        g00 = wmma4(a0, p0, g00);
        g01 = wmma4(a0, p1, g01);
        g10 = wmma4(a1, p0, g10);
        g11 = wmma4(a1, p1, g11);
        u00 = wmma4(a0, q0, u00);
        u01 = wmma4(a0, q1, u01);
        u10 = wmma4(a1, q0, u10);
        u11 = wmma4(a1, q1, u11);
        a0 = na0; a1 = na1;
        p0 = np0; p1 = np1; q0 = nq0; q1 = nq1;
    }
    g00 = wmma4(a0, p0, g00);
    g01 = wmma4(a0, p1, g01);
    g10 = wmma4(a1, p0, g10);
    g11 = wmma4(a1, p1, g11);
    u00 = wmma4(a0, q0, u00);
    u01 = wmma4(a0, q1, u01);
    u10 = wmma4(a1, q0, u10);
    u11 = wmma4(a1, q1, u11);

#pragma unroll
    for (int v = 0; v < 8; ++v) {
        int r0 = mbase + v + 8 * hi;
        int r1 = r0 + 16;
        size_t base0 = (size_t)r0 * N + nbase + lo;
        size_t base1 = (size_t)r1 * N + nbase + lo;
        float av, gv, sig;
        av = g00[v]; gv = u00[v]; sig = 1.0f / (1.0f + __expf(-av));
        H[base0] = av * sig * gv;
        av = g01[v]; gv = u01[v]; sig = 1.0f / (1.0f + __expf(-av));
        H[base0 + 16] = av * sig * gv;
        av = g10[v]; gv = u10[v]; sig = 1.0f / (1.0f + __expf(-av));
        H[base1] = av * sig * gv;
        av = g11[v]; gv = u11[v]; sig = 1.0f / (1.0f + __expf(-av));
        H[base1 + 16] = av * sig * gv;
    }
}

// ---- Flash attention: one wave per 16-query tile per (b,h) ------------------
__global__ void attention_kernel(const float* __restrict__ Q,
                                 const float* __restrict__ Kg,
                                 const float* __restrict__ V,
                                 float* __restrict__ O) {
    __shared__ float pshare[8][16][17];   // per-wave P tile, padded
    int wave = threadIdx.x >> 5;
    int lane = threadIdx.x & 31;
    int hi = lane >> 4, lo = lane & 15;

    int id = blockIdx.x * 8 + wave;       // 4096 ids = 2 * 16 * 128
    int qt = id & 127;
    int h  = (id >> 7) & 15;
    int b  = id >> 11;

    size_t hoff = (size_t)b * SEQ * D_MODEL + (size_t)h * D_K;
    const float* Qb = Q + hoff;
    const float* Kb = Kg + hoff;
    const float* Vb = V + hoff;
    float* Ob = O + hoff;

    int qbase = qt * 16;

    // Q tile as 16 A-fragments (d = 0..63 in steps of 4), lives in registers
    v2f qa[16];
    const float* qrow = Qb + (size_t)(qbase + lo) * D_MODEL + 2 * hi;
#pragma unroll
    for (int t = 0; t < 16; ++t) qa[t] = *(const v2f*)(qrow + 4 * t);

    v8f acc0 = {}, acc1 = {}, acc2 = {}, acc3 = {};
    float run_m[8], run_l[8];
#pragma unroll
    for (int v = 0; v < 8; ++v) { run_m[v] = -1e30f; run_l[v] = 0.0f; }

    for (int j = 0; j <= qt; ++j) {
        int jbase = j * 16;
        // S tile = Q @ K^T (16 fp32 WMMAs)
        v8f s = {};
        const float* krow = Kb + (size_t)(jbase + lo) * D_MODEL + 2 * hi;
#pragma unroll
        for (int t = 0; t < 16; ++t) {
            v2f kb = *(const v2f*)(krow + 4 * t);
            s = wmma4(qa[t], kb, s);
        }
        // online softmax; each row m = v + 8*hi spans 16 lanes of fixed hi
        float ps[8];
#pragma unroll
        for (int v = 0; v < 8; ++v) {
            float sv = s[v] * 0.125f;                 // 1/sqrt(64)
            int qi = qbase + v + 8 * hi;
            int ki = jbase + lo;
            if (ki > qi) sv = -1e30f;                 // causal mask
            float rmax = sv;
#pragma unroll
            for (int m = 1; m < 16; m <<= 1) rmax = fmaxf(rmax, __shfl_xor(rmax, m, 32));
            float mnew = fmaxf(run_m[v], rmax);
            float alpha = __expf(run_m[v] - mnew);
            float p = __expf(sv - mnew);
            float psum = p;
#pragma unroll
            for (int m = 1; m < 16; m <<= 1) psum += __shfl_xor(psum, m, 32);
            run_l[v] = run_l[v] * alpha + psum;
            run_m[v] = mnew;
            acc0[v] *= alpha; acc1[v] *= alpha; acc2[v] *= alpha; acc3[v] *= alpha;
            ps[v] = p;
        }
        // transpose P (C-layout -> A-layout) through per-wave LDS tile;
        // same-wave DS ops are in order, no barrier needed
#pragma unroll
        for (int v = 0; v < 8; ++v) pshare[wave][v + 8 * hi][lo] = ps[v];
        v2f pa[4];
#pragma unroll
        for (int t = 0; t < 4; ++t) {
            pa[t].x = pshare[wave][lo][4 * t + 2 * hi];
            pa[t].y = pshare[wave][lo][4 * t + 2 * hi + 1];
        }
        // ctx += P @ V (16 fp32 WMMAs across 4 d-tiles)
#pragma unroll
        for (int t = 0; t < 4; ++t) {
            const float* vr0 = Vb + (size_t)(jbase + 4 * t + 2 * hi) * D_MODEL + lo;
            const float* vr1 = vr0 + D_MODEL;
            v2f vb0 = { vr0[0],  vr1[0]  };
            v2f vb1 = { vr0[16], vr1[16] };
            v2f vb2 = { vr0[32], vr1[32] };
            v2f vb3 = { vr0[48], vr1[48] };
            acc0 = wmma4(pa[t], vb0, acc0);
            acc1 = wmma4(pa[t], vb1, acc1);
            acc2 = wmma4(pa[t], vb2, acc2);
            acc3 = wmma4(pa[t], vb3, acc3);
        }
    }
    // normalize and store ctx in (b, s, h*64+d) layout
#pragma unroll
    for (int v = 0; v < 8; ++v) {
        float inv = 1.0f / run_l[v];
        int row = qbase + v + 8 * hi;
        float* orow = Ob + (size_t)row * D_MODEL + lo;
        orow[0]  = acc0[v] * inv;
        orow[16] = acc1[v] * inv;
        orow[32] = acc2[v] * inv;
        orow[48] = acc3[v] * inv;
    }
}

// -----------------------------------------------------------------------------
extern "C" void kernel_launch(void* const* d_in, const int* in_sizes, int n_in,
                              void* d_out, int out_size, void* d_ws, size_t ws_size,
                              hipStream_t stream) {
    (void)in_sizes; (void)n_in; (void)out_size; (void)ws_size;
    const float* x   = (const float*)d_in[0];
    const float* Wq  = (const float*)d_in[1];
    const float* Wk  = (const float*)d_in[2];
    const float* Wv  = (const float*)d_in[3];
    const float* Wo  = (const float*)d_in[4];
    const float* ln1 = (const float*)d_in[5];
    const float* ln2 = (const float*)d_in[6];
    const float* w1  = (const float*)d_in[7];
    const float* w2  = (const float*)d_in[8];
    const float* w3  = (const float*)d_in[9];
    float* out = (float*)d_out;

    float* ws = (float*)d_ws;
    float* XN = ws;                      // 16 MB, reused: xn1 -> ctx -> xn2
    float* Qb = ws + 1 * BUF_ELEMS;      // 16 MB
    float* Kb = ws + 2 * BUF_ELEMS;      // 16 MB
    float* Vb = ws + 3 * BUF_ELEMS;      // 16 MB
    float* Hb = ws + 4 * BUF_ELEMS;      // 64 MB (M_TOT x D_FF)

    dim3 b256(256);
    dim3 gP(M_TOT / 64, D_MODEL / 256);  // 64 x 4   (proj / down GEMMs, N=1024)
    dim3 gF(M_TOT / 64, D_FF / 128);     // 64 x 32  (fused swiglu GEMM, N=4096)

    // 1. xn = rmsnorm(x, ln1)
    rmsnorm_kernel<<<M_TOT, b256, 0, stream>>>(x, ln1, XN);
    // 2. Q,K,V projections
    gemm_nt_kernel<0><<<gP, b256, 0, stream>>>(XN, Wq, Qb, nullptr, M_TOT, D_MODEL, D_MODEL);
    gemm_nt_kernel<0><<<gP, b256, 0, stream>>>(XN, Wk, Kb, nullptr, M_TOT, D_MODEL, D_MODEL);
    gemm_nt_kernel<0><<<gP, b256, 0, stream>>>(XN, Wv, Vb, nullptr, M_TOT, D_MODEL, D_MODEL);
    // 3. RoPE on Q, K
    int npairs = M_TOT * (D_MODEL / 2);
    rope_kernel<<<(npairs + 255) / 256, b256, 0, stream>>>(Qb, npairs);
    rope_kernel<<<(npairs + 255) / 256, b256, 0, stream>>>(Kb, npairs);
    // 4. flash attention -> ctx (reuse XN)
    attention_kernel<<<512, b256, 0, stream>>>(Qb, Kb, Vb, XN);
    // 5. x1 = x + ctx @ Wo^T  -> out
    gemm_nt_kernel<1><<<gP, b256, 0, stream>>>(XN, Wo, out, x, M_TOT, D_MODEL, D_MODEL);
    // 6. xn2 = rmsnorm(x1, ln2) -> XN
    rmsnorm_kernel<<<M_TOT, b256, 0, stream>>>(out, ln2, XN);
    // 7. h = silu(xn2@w1^T) * (xn2@w3^T) -> Hb
    gemm_swiglu_kernel<<<gF, b256, 0, stream>>>(XN, w1, w3, Hb, M_TOT, D_FF, D_MODEL);
    // 8. out = x1 + h @ w2^T
    gemm_nt_kernel<1><<<gP, b256, 0, stream>>>(Hb, w2, out, out, M_TOT, D_MODEL, D_FF);
}